// GINWithNoise_69441031242294
// MI455X (gfx1250) — compile-verified
//
#include <hip/hip_runtime.h>

// ---------------------------------------------------------------------------
// GIN with noise, CDNA5 (gfx1250).
//   h0 = concat(X, eps)                       [S*N, 192]
//   agg = scatter_add(h[src] -> dst)          (global f32 atomics, L2-resident)
//   h1  = relu(((1+e0)*h0 + agg0) @ W0 + b0)  (V_WMMA_F32_16X16X4_F32)
//   out = relu(((1+e1)*h1 + agg1) @ W1 + b1)
// Output tuple = (out [S*N*256], epsilon [S*N*64]) concatenated flat.
// ---------------------------------------------------------------------------

typedef __attribute__((ext_vector_type(2))) float v2f;
typedef __attribute__((ext_vector_type(8))) float v8f;

#define N_NODES 50000
#define N_EDGES 800000
#define S_SAMP  2
#define D_IN    128
#define D_NOISE 64
#define D_H0    (D_IN + D_NOISE)      // 192
#define D_L0    256
#define D_L1    256
#define SN      (S_SAMP * N_NODES)    // 100000 rows (divisible by 16)

// ---------------------------------------------------------------- utilities
__global__ void zero_kernel(float* __restrict__ p, long long n) {
  long long tid = (long long)blockIdx.x * blockDim.x + threadIdx.x;
  if (tid < n) p[tid] = 0.0f;
}

__global__ void concat_kernel(const float* __restrict__ X,    // [N, 128]
                              const float* __restrict__ eps,  // [S, N, 64]
                              float* __restrict__ h0) {       // [S*N, 192]
  int tid = blockIdx.x * blockDim.x + threadIdx.x;
  const int total = SN * D_H0;
  if (tid >= total) return;
  int r = tid / D_H0;          // row = s*N + n
  int k = tid - r * D_H0;
  int n = r % N_NODES;
  float v;
  if (k < D_IN) v = X[n * D_IN + k];
  else          v = eps[(long long)r * D_NOISE + (k - D_IN)];
  h0[tid] = v;
}

__global__ void copy_eps_kernel(const float* __restrict__ eps,
                                float* __restrict__ out) {
  int tid = blockIdx.x * blockDim.x + threadIdx.x;
  const int total = SN * D_NOISE;
  if (tid < total) out[tid] = eps[tid];
}

// ------------------------------------------------------------- scatter-add
// One thread per (edge, feature); k is fastest-varying -> waves touch 32
// consecutive dwords of one source/dest row (coalesced). Both samples done by
// the same thread (edge index scalar-cached). agg rows: sample s at s*N + node.
template<int D>
__global__ void scatter_add_kernel(const float* __restrict__ H,     // [S*N, D]
                                   const int* __restrict__ src,
                                   const int* __restrict__ dst,
                                   float* __restrict__ agg) {       // [S*N, D]
  long long tid = (long long)blockIdx.x * blockDim.x + threadIdx.x;
  const long long total = (long long)N_EDGES * D;
  if (tid >= total) return;
  int e = (int)(tid / D);
  int k = (int)(tid - (long long)e * D);
  int s = src[e];
  int d = dst[e];
  atomicAdd(&agg[(long long)d * D + k],              H[(long long)s * D + k]);
  atomicAdd(&agg[((long long)N_NODES + d) * D + k],  H[((long long)N_NODES + s) * D + k]);
}

// --------------------------------------------------------------- WMMA GEMM
// D = A x B + C with V_WMMA_F32_16X16X4_F32 (keeps fp32 semantics).
// Fragment layouts (ISA 7.12.2, 32-bit):
//   A 16x4 : lane = 16*(K>=2) + M ; VGPR v holds K = 2*(lane>>4) + v
//   B 4x16 : lane = 16*(K-half) + N ; VGPR v holds K = 2*(lane>>4) + v
//   C 16x16: VGPR v, lanes 0-15 -> M=v, lanes 16-31 -> M=v+8 ; N = lane&15
// Each wave computes a 16(M) x 64(N) strip: one A fragment reused by 4 B tiles.
// A operand is built on the fly: z = (1+eps)*Hin + Agg.
template<int K, int DN>
__global__ void gin_gemm_wmma(const float* __restrict__ Hin,   // [SN, K]
                              const float* __restrict__ Agg,   // [SN, K]
                              const float* __restrict__ W,     // [K, DN]
                              const float* __restrict__ bias,  // [DN]
                              const float* __restrict__ epsv,  // scalar
                              float* __restrict__ Hout) {      // [SN, DN]
  constexpr int NSTRIP = DN / 64;
  const int lane = threadIdx.x & 31;
  const int wave = threadIdx.x >> 5;
  const int gw   = blockIdx.x * (blockDim.x >> 5) + wave;
  const int mt   = gw / NSTRIP;           // M tile (16 rows)
  const int ns   = gw - mt * NSTRIP;      // 64-col strip
  if (mt * 16 >= SN) return;

  const int lo = lane & 15;
  const int hi = lane >> 4;               // 0 or 1: selects K pair
  const int row = mt * 16 + lo;           // A row this lane supplies
  const int col0 = ns * 64 + lo;          // first B/C column this lane owns
  const float onepe = 1.0f + epsv[0];

  v8f acc[4];
#pragma unroll
  for (int t = 0; t < 4; ++t) acc[t] = (v8f){0.f,0.f,0.f,0.f,0.f,0.f,0.f,0.f};

#pragma unroll 4
  for (int k0 = 0; k0 < K; k0 += 4) {
    const int ka = k0 + 2 * hi;           // this lane's first K index
    const float2 h2 = *reinterpret_cast<const float2*>(Hin + (long long)row * K + ka);
    const float2 g2 = *reinterpret_cast<const float2*>(Agg + (long long)row * K + ka);
    v2f a;
    a.x = onepe * h2.x + g2.x;
    a.y = onepe * h2.y + g2.y;

    const float* wr0 = W + (long long)ka * DN + col0;  // row K=ka
    const float* wr1 = wr0 + DN;                       // row K=ka+1
#pragma unroll
    for (int t = 0; t < 4; ++t) {
      v2f b;
      b.x = wr0[16 * t];
      b.y = wr1[16 * t];
      acc[t] = __builtin_amdgcn_wmma_f32_16x16x4_f32(
          /*neg_a=*/false, a, /*neg_b=*/false, b,
          /*c_mod=*/(short)0, acc[t], /*reuse_a=*/false, /*reuse_b=*/false);
    }
  }

  // Epilogue: bias + ReLU, coalesced 16-dword segments per (t, v).
#pragma unroll
  for (int t = 0; t < 4; ++t) {
    const int col = col0 + 16 * t;
    const float bb = bias[col];
#pragma unroll
    for (int v = 0; v < 8; ++v) {
      const int ro = mt * 16 + v + 8 * hi;
      float val = acc[t][v] + bb;
      val = val > 0.0f ? val : 0.0f;
      Hout[(long long)ro * DN + col] = val;
    }
  }
}

// ---------------------------------------------------------------------------
extern "C" void kernel_launch(void* const* d_in, const int* in_sizes, int n_in,
                              void* d_out, int out_size, void* d_ws, size_t ws_size,
                              hipStream_t stream) {
  const float* X    = (const float*)d_in[0];
  const float* eps  = (const float*)d_in[1];
  const int*   esrc = (const int*)d_in[2];
  const int*   edst = (const int*)d_in[3];
  const float* W0   = (const float*)d_in[4];
  const float* b0   = (const float*)d_in[5];
  const float* W1   = (const float*)d_in[6];
  const float* b1   = (const float*)d_in[7];
  const float* e0   = (const float*)d_in[8];
  const float* e1   = (const float*)d_in[9];

  // Workspace layout (floats). agg1 reuses the h0+agg0 region (free by then).
  //   h0   : [0,               SN*192)
  //   agg0 : [SN*192,          2*SN*192)
  //   h1   : [2*SN*192,        2*SN*192 + SN*256)
  //   agg1 : [0,               SN*256)
  // Total required: (2*192 + 256) * SN floats = 256 MB.
  const size_t need = (size_t)(2 * D_H0 + D_L0) * SN * sizeof(float);
  if (ws_size < need) return;  // harness must provide >= 256 MB scratch

  float* ws   = (float*)d_ws;
  float* h0   = ws;
  float* agg0 = ws + (size_t)SN * D_H0;
  float* h1   = ws + (size_t)2 * SN * D_H0;
  float* agg1 = ws;

  float* outF    = (float*)d_out;
  float* out_eps = outF + (size_t)SN * D_L1;

  const int B = 256;  // 8 wave32 per block

  // 1) h0 = concat(X, eps)
  concat_kernel<<<(SN * D_H0 + B - 1) / B, B, 0, stream>>>(X, eps, h0);

  // 2) layer 0 aggregation
  zero_kernel<<<(SN * D_H0 + B - 1) / B, B, 0, stream>>>(agg0, (long long)SN * D_H0);
  {
    long long tot = (long long)N_EDGES * D_H0;
    scatter_add_kernel<D_H0><<<(unsigned)((tot + B - 1) / B), B, 0, stream>>>(h0, esrc, edst, agg0);
  }

  // 3) layer 0 GEMM + bias + ReLU -> h1
  {
    int nwaves = (SN / 16) * (D_L0 / 64);  // 25000
    gin_gemm_wmma<D_H0, D_L0><<<nwaves / 8, B, 0, stream>>>(h0, agg0, W0, b0, e0, h1);
  }

  // 4) layer 1 aggregation (agg1 overlaps dead h0/agg0 region)
  zero_kernel<<<(SN * D_L0 + B - 1) / B, B, 0, stream>>>(agg1, (long long)SN * D_L0);
  {
    long long tot = (long long)N_EDGES * D_L0;
    scatter_add_kernel<D_L0><<<(unsigned)((tot + B - 1) / B), B, 0, stream>>>(h1, esrc, edst, agg1);
  }

  // 5) layer 1 GEMM + bias + ReLU -> d_out (outer relu is idempotent)
  {
    int nwaves = (SN / 16) * (D_L1 / 64);
    gin_gemm_wmma<D_L0, D_L1><<<nwaves / 8, B, 0, stream>>>(h1, agg1, W1, b1, e1, outF);
  }

  // 6) tuple tail: epsilon passthrough
  copy_eps_kernel<<<(SN * D_NOISE + B - 1) / B, B, 0, stream>>>(eps, out_eps);
}